// RoiPoolingConv_25202868093668
// MI455X (gfx1250) — compile-verified
//
#include <hip/hip_runtime.h>
#include <stdint.h>

#define POOL   7
#define NROIS  1024
#define IMG_H  256
#define IMG_W  256
#define CH     512

typedef float v2f __attribute__((ext_vector_type(2)));
typedef int   v4i __attribute__((vector_size(16)));   // matches builtin's pointee type

#define AS1 __attribute__((address_space(1)))
#define AS3 __attribute__((address_space(3)))

// ---- CDNA5 async global->LDS copy (ASYNCcnt-tracked) -----------------------
__device__ __forceinline__ void async_copy_b128(const float* gsrc, float* ldst) {
#if __has_builtin(__builtin_amdgcn_global_load_async_to_lds_b128)
  __builtin_amdgcn_global_load_async_to_lds_b128(
      (AS1 v4i*)(uintptr_t)gsrc,   // global src (flat addr == AS1 addr on amdgcn)
      (AS3 v4i*)ldst,              // addrspacecast generic->LDS
      /*offset=*/0, /*cpol=*/0);
#else
  asm volatile("global_load_async_to_lds_b128 %0, %1, off"
               :
               : "v"((uint32_t)(uintptr_t)(AS3 float*)ldst),
                 "v"((AS1 const void*)gsrc)
               : "memory");
#endif
}

template <int N>
__device__ __forceinline__ void wait_async() {
#if __has_builtin(__builtin_amdgcn_s_wait_asynccnt)
  __builtin_amdgcn_s_wait_asynccnt(N);
#else
  if constexpr (N == 0)
    asm volatile("s_wait_asynccnt 0x0" ::: "memory");
  else
    asm volatile("s_wait_asynccnt 0x2" ::: "memory");
#endif
}

// ---------------------------------------------------------------------------
// One block per (roi, py): 7 output pixels x 512 channels.
// LDS double buffer: [2 stages][4 corners][512 floats] = 16 KB.
// Pipeline: async-stage corners for px+1 while blending px from LDS.
// ---------------------------------------------------------------------------
__global__ void __launch_bounds__(256)
roi_pool_kernel(const float* __restrict__ feat,
                const int*   __restrict__ rois,
                float*       __restrict__ out) {
  __shared__ float lds[2][4][CH];

  const int block = blockIdx.x;           // roi * POOL + py
  const int roi   = block / POOL;
  const int py    = block - roi * POOL;
  const int t     = threadIdx.x;          // 0..255

  const int x1 = rois[roi * 4 + 0];
  const int y1 = rois[roi * 4 + 1];
  const int x2 = rois[roi * 4 + 2];
  const int y2 = rois[roi * 4 + 3];

  // Reference mapping (TF1 legacy): ys = py * (h/P), y0 = floor,
  // y1i = min(y0+1, max(h-1,0)), wy = ys - y0.  (rows fixed per block)
  const float sy  = (float)(y2 - y1) / (float)POOL;
  const float sx  = (float)(x2 - x1) / (float)POOL;
  const float ysf = (float)py * sy;
  const int   y0  = (int)ysf;             // ysf >= 0 -> trunc == floor
  const float wy  = ysf - (float)y0;
  const int   hm1 = max(y2 - y1 - 1, 0);
  const int   wm1 = max(x2 - x1 - 1, 0);
  const int   r0  = y1 + y0;
  const int   r1  = y1 + min(y0 + 1, hm1);

  auto cols = [&](int px, int& c0, int& c1, float& wx) {
    const float xsf = (float)px * sx;
    const int   x0  = (int)xsf;
    wx = xsf - (float)x0;
    c0 = x1 + x0;
    c1 = x1 + min(x0 + 1, wm1);
  };

  // Stage one pixel's 4 corner rows (4 x 2KB) into LDS: 512 b128 units,
  // 2 per thread.  unit u: corner = u>>7 (00,01,10,11), q = u&127 (16B unit).
  auto issue = [&](int buf, int c0, int c1) {
#pragma unroll
    for (int s = 0; s < 2; ++s) {
      const int u      = t + s * 256;
      const int corner = u >> 7;
      const int q      = u & 127;
      const int row    = (corner < 2) ? r0 : r1;
      const int col    = (corner & 1) ? c1 : c0;
      const float* g   = feat + ((size_t)(row * IMG_W + col)) * CH + (size_t)q * 4;
      async_copy_b128(g, &lds[buf][corner][q * 4]);
    }
  };

  int   c0 = 0, c1 = 0;
  float wx_cur = 0.0f;
  cols(0, c0, c1, wx_cur);
  issue(0, c0, c1);

  const int cbase = t * 2;                // two channels per thread

  for (int px = 0; px < POOL; ++px) {
    int   nc0 = 0, nc1 = 0;
    float nwx = 0.0f;
    if (px + 1 < POOL) {
      cols(px + 1, nc0, nc1, nwx);
      issue((px + 1) & 1, nc0, nc1);      // prefetch next pixel
      wait_async<2>();                    // drain px's stage (2 outstanding = next)
    } else {
      wait_async<0>();
    }
    __syncthreads();                      // all waves' stage-px data in LDS

    const int  b    = px & 1;
    const v2f  v00  = *(const v2f*)&lds[b][0][cbase];
    const v2f  v01  = *(const v2f*)&lds[b][1][cbase];
    const v2f  v10  = *(const v2f*)&lds[b][2][cbase];
    const v2f  v11  = *(const v2f*)&lds[b][3][cbase];
    const float omy = 1.0f - wy;
    const float omx = 1.0f - wx_cur;
    const v2f  r    = v00 * (omy * omx) + v01 * (omy * wx_cur) +
                      v10 * (wy  * omx) + v11 * (wy  * wx_cur);

    // Streamed 103 MB output: non-temporal so the L2-resident image survives.
    float* o = out + ((size_t)block * POOL + px) * CH + cbase;
    __builtin_nontemporal_store(r, (v2f*)o);

    __syncthreads();                      // reads done before buffer is reused
    wx_cur = nwx;
  }
}

extern "C" void kernel_launch(void* const* d_in, const int* in_sizes, int n_in,
                              void* d_out, int out_size, void* d_ws, size_t ws_size,
                              hipStream_t stream) {
  const float* img  = (const float*)d_in[0];   // [1,256,256,512] f32
  const int*   rois = (const int*)d_in[1];     // [1,1024,4] i32 (x1,y1,x2,y2)
  float*       out  = (float*)d_out;           // [1,1024,7,7,512] f32

  dim3 grid(NROIS * POOL);
  dim3 block(256);
  roi_pool_kernel<<<grid, block, 0, stream>>>(img, rois, out);
}